// ImprovedGNNEncoder_73555609911548
// MI455X (gfx1250) — compile-verified
//
#include <hip/hip_runtime.h>
#include <stdint.h>

#define N_NODES 100000
#define N_EDGES 1600000
#define D 128
#define N_LAYERS 4
#define LN_EPS 1e-5f

typedef __attribute__((ext_vector_type(2))) float v2f;
typedef __attribute__((ext_vector_type(8))) float v8f;

// ---------------------------------------------------------------------------
// Utility: grid-stride zero fill
// ---------------------------------------------------------------------------
__global__ void zero_kernel(float* __restrict__ p, long long n) {
    long long i = (long long)blockIdx.x * blockDim.x + threadIdx.x;
    long long stride = (long long)gridDim.x * blockDim.x;
    for (; i < n; i += stride) p[i] = 0.0f;
}

// ---------------------------------------------------------------------------
// Degree: atomicAdd 1.0 on dst per edge
// ---------------------------------------------------------------------------
__global__ void deg_kernel(const long long* __restrict__ dst, float* __restrict__ deg,
                           long long E) {
    long long e = (long long)blockIdx.x * blockDim.x + threadIdx.x;
    if (e >= E) return;
    atomicAdd(&deg[dst[e]], 1.0f);
}

// deg -> dinv in place: dinv = rsqrt(deg + 1)
__global__ void dinv_kernel(float* __restrict__ deg, int n) {
    int i = blockIdx.x * blockDim.x + threadIdx.x;
    if (i >= n) return;
    deg[i] = rsqrtf(deg[i] + 1.0f);
}

// ---------------------------------------------------------------------------
// WMMA f32 GEMM: Out[N x 128] = H[N x 128] @ W[128 x 128]
// One wave per 16-row strip. 8 column tiles of 16, K in steps of 4 using
// V_WMMA_F32_16X16X4_F32.
//
// ISA layouts (CDNA5 ISA):
//  A 16x4 f32 : lane l: m = l&15;  vgpr v holds K = 2*(l>>4) + v
//  B 4x16 f32 : lane l: n = l&15;  vgpr v holds K = 2*(l>>4) + v   (mirror of A)
//  C/D 16x16  : vgpr r: lanes0-15 -> M=r, N=lane; lanes16-31 -> M=r+8, N=lane-16
// ---------------------------------------------------------------------------
__global__ void gemm_wmma_kernel(const float* __restrict__ H,
                                 const float* __restrict__ W,
                                 float* __restrict__ Out,
                                 int nStrips) {
    const int wavesPerBlock = blockDim.x >> 5;
    const int strip = blockIdx.x * wavesPerBlock + (threadIdx.x >> 5);
    if (strip >= nStrips) return;                 // wave-uniform exit (EXEC stays full)
    const int lane  = threadIdx.x & 31;
    const int m     = lane & 15;
    const int khalf = lane >> 4;                  // 0 or 1
    const int row0  = strip * 16;
    const int row   = row0 + m;

    v8f acc[8];
#pragma unroll
    for (int t = 0; t < 8; ++t) acc[t] = (v8f){0, 0, 0, 0, 0, 0, 0, 0};

    const float* arow = H + (size_t)row * D;
    for (int kb = 0; kb < D; kb += 4) {
        v2f a;
        {
            const float* ap = arow + kb + khalf * 2;
            a.x = ap[0];
            a.y = ap[1];
        }
#pragma unroll
        for (int t = 0; t < 8; ++t) {
            const int n = t * 16 + m;
            v2f b;
            const float* bp = W + (size_t)(kb + khalf * 2) * D + n;
            b.x = bp[0];
            b.y = bp[D];
            acc[t] = __builtin_amdgcn_wmma_f32_16x16x4_f32(
                /*neg_a=*/false, a, /*neg_b=*/false, b,
                /*c_mod=*/(short)0, acc[t],
                /*reuse_a=*/false, /*reuse_b=*/false);
        }
    }

    // Store D-matrix
    const int colb = lane & 15;
#pragma unroll
    for (int t = 0; t < 8; ++t) {
#pragma unroll
        for (int r = 0; r < 8; ++r) {
            const int orow = row0 + r + khalf * 8;
            Out[(size_t)orow * D + t * 16 + colb] = acc[t][r];
        }
    }
}

// ---------------------------------------------------------------------------
// Edge scatter: agg[dst] += hW[src] * dinv[src]*dinv[dst]
// One wave per edge; each lane handles 4 consecutive floats (float4).
// ---------------------------------------------------------------------------
__global__ void scatter_kernel(const long long* __restrict__ src,
                               const long long* __restrict__ dst,
                               const float* __restrict__ dinv,
                               const float* __restrict__ hW,
                               float* __restrict__ agg,
                               long long E) {
    long long gid = (long long)blockIdx.x * blockDim.x + threadIdx.x;
    long long e = gid >> 5;
    int lane = (int)(gid & 31);
    if (e >= E) return;
    long long s = src[e];
    long long d = dst[e];
    float norm = dinv[s] * dinv[d];
    float4 v = ((const float4*)(hW + (size_t)s * D))[lane];
    float* ap = agg + (size_t)d * D + lane * 4;
    atomicAdd(ap + 0, v.x * norm);
    atomicAdd(ap + 1, v.y * norm);
    atomicAdd(ap + 2, v.z * norm);
    atomicAdd(ap + 3, v.w * norm);
}

// ---------------------------------------------------------------------------
// Post: t = agg + hW*dinv^2 + b ; LayerNorm(gamma,beta) ; ReLU
// One wave per node; 32 lanes x float4 = 128 features. Wave32 shuffles.
// ---------------------------------------------------------------------------
__global__ void post_kernel(const float* __restrict__ agg,
                            const float* __restrict__ hW,
                            const float* __restrict__ dinv,
                            const float* __restrict__ b,
                            const float* __restrict__ gamma,
                            const float* __restrict__ beta,
                            float* __restrict__ out,
                            int nNodes) {
    int node = blockIdx.x * (blockDim.x >> 5) + (threadIdx.x >> 5);
    int lane = threadIdx.x & 31;
    if (node >= nNodes) return;

    float di = dinv[node];
    float di2 = di * di;
    float4 av = ((const float4*)(agg + (size_t)node * D))[lane];
    float4 hv = ((const float4*)(hW + (size_t)node * D))[lane];
    float4 bv = ((const float4*)b)[lane];

    float4 t;
    t.x = av.x + hv.x * di2 + bv.x;
    t.y = av.y + hv.y * di2 + bv.y;
    t.z = av.z + hv.z * di2 + bv.z;
    t.w = av.w + hv.w * di2 + bv.w;

    // mean
    float sum = t.x + t.y + t.z + t.w;
#pragma unroll
    for (int off = 16; off > 0; off >>= 1) sum += __shfl_xor(sum, off, 32);
    float mu = sum * (1.0f / (float)D);

    float dx0 = t.x - mu, dx1 = t.y - mu, dx2 = t.z - mu, dx3 = t.w - mu;
    float vs = dx0 * dx0 + dx1 * dx1 + dx2 * dx2 + dx3 * dx3;
#pragma unroll
    for (int off = 16; off > 0; off >>= 1) vs += __shfl_xor(vs, off, 32);
    float var = vs * (1.0f / (float)D);
    float rinv = rsqrtf(var + LN_EPS);

    float4 gv = ((const float4*)gamma)[lane];
    float4 bev = ((const float4*)beta)[lane];
    float4 o;
    o.x = fmaxf(dx0 * rinv * gv.x + bev.x, 0.0f);
    o.y = fmaxf(dx1 * rinv * gv.y + bev.y, 0.0f);
    o.z = fmaxf(dx2 * rinv * gv.z + bev.z, 0.0f);
    o.w = fmaxf(dx3 * rinv * gv.w + bev.w, 0.0f);
    ((float4*)(out + (size_t)node * D))[lane] = o;
}

// ---------------------------------------------------------------------------
// Launch
// ---------------------------------------------------------------------------
extern "C" void kernel_launch(void* const* d_in, const int* in_sizes, int n_in,
                              void* d_out, int out_size, void* d_ws, size_t ws_size,
                              hipStream_t stream) {
    (void)in_sizes; (void)n_in; (void)out_size; (void)ws_size;

    const float*     x      = (const float*)d_in[0];
    const long long* ei     = (const long long*)d_in[1];   // int64 [2, E]
    const float*     Ws     = (const float*)d_in[2];       // [4, D, D]
    const float*     bs     = (const float*)d_in[3];       // [4, D]
    const float*     gammas = (const float*)d_in[4];       // [4, D]
    const float*     betas  = (const float*)d_in[5];       // [4, D]
    float*           out    = (float*)d_out;               // [N, D]

    const long long* src = ei;
    const long long* dstp = ei + N_EDGES;

    // Workspace carve-up: deg/dinv [N], hW [N*D], agg [N*D]
    float* deg = (float*)d_ws;                      // becomes dinv in place
    float* hW  = deg + N_NODES;
    float* agg = hW + (size_t)N_NODES * D;

    const long long nd = (long long)N_NODES * D;

    // --- degree / dinv (once) ---
    zero_kernel<<<1024, 256, 0, stream>>>(deg, N_NODES);
    deg_kernel<<<(N_EDGES + 255) / 256, 256, 0, stream>>>(dstp, deg, N_EDGES);
    dinv_kernel<<<(N_NODES + 255) / 256, 256, 0, stream>>>(deg, N_NODES);
    float* dinv = deg;

    const int nStrips = N_NODES / 16;               // 6250 (N divisible by 16)
    const int gemmBlocks = (nStrips + 3) / 4;       // 4 waves of 32 per block
    const long long scatterThreads = (long long)N_EDGES * 32;
    const int scatterBlocks = (int)((scatterThreads + 255) / 256);
    const int postBlocks = (N_NODES + 7) / 8;       // 8 waves of 32 per block

    for (int layer = 0; layer < N_LAYERS; ++layer) {
        const float* hin = (layer == 0) ? x : out;  // in-place across layers is safe:
                                                    // post reads only agg/hW
        // 1) hW = hin @ W[layer]
        gemm_wmma_kernel<<<gemmBlocks, 128, 0, stream>>>(
            hin, Ws + (size_t)layer * D * D, hW, nStrips);
        // 2) agg = 0
        zero_kernel<<<4096, 256, 0, stream>>>(agg, nd);
        // 3) agg[dst] += hW[src] * dinv[src]*dinv[dst]
        scatter_kernel<<<scatterBlocks, 256, 0, stream>>>(src, dstp, dinv, hW, agg, N_EDGES);
        // 4) out = relu(LN(agg + hW*dinv^2 + b))
        post_kernel<<<postBlocks, 256, 0, stream>>>(
            agg, hW, dinv, bs + (size_t)layer * D,
            gammas + (size_t)layer * D, betas + (size_t)layer * D,
            out, N_NODES);
    }
}